// ButterflyLinear_70617852280893
// MI455X (gfx1250) — compile-verified
//
#include <hip/hip_runtime.h>

typedef __attribute__((ext_vector_type(2))) float v2f;
typedef __attribute__((ext_vector_type(4))) float v4f;
typedef __attribute__((ext_vector_type(8))) float v8f;

#define BN      4096
#define BTOK    8192
#define BDEPTH  12
#define NGROUP  1024            // BN/4
#define LDS_PITCH 132           // floats; 528B row pitch: 16B aligned, conflict-free

// ---------------------------------------------------------------------------
// Pass 1: fold the 12 butterfly stages into 1024 per-group 4x4 matrices W_g.
// (All stages only mix within aligned 4-column groups: stage 0 pairs (j,j+1),
//  stages >=1 pair (j,j+2) inside the group.)  Row-vector: out4 = in4 @ W_g.
// factors layout: factors[s][p][c][d] flat = ((s*2048 + p)*2 + c)*2 + d
// ---------------------------------------------------------------------------
__global__ __launch_bounds__(256) void bfly_fold_kernel(
    const float* __restrict__ factors, float* __restrict__ Wfold) {
  int g = blockIdx.x * blockDim.x + threadIdx.x;
  if (g >= NGROUP) return;

  float w[4][4];
  const float* F0 = factors + (size_t)(2 * g) * 4;   // rows 2g,2g+1: 8 floats
  #pragma unroll
  for (int i = 0; i < 4; ++i)
    #pragma unroll
    for (int j = 0; j < 4; ++j) w[i][j] = 0.0f;
  w[0][0] = F0[0]; w[0][1] = F0[1];
  w[1][0] = F0[2]; w[1][1] = F0[3];
  w[2][2] = F0[4]; w[2][3] = F0[5];
  w[3][2] = F0[6]; w[3][3] = F0[7];

  for (int s = 1; s < BDEPTH; ++s) {
    unsigned j0   = 4u * (unsigned)g;
    unsigned mask = (2u << s) - 1u;                      // 2^(s+1)-1
    unsigned p0   = ((j0 >> (s + 1)) << s) | ((j0 & mask) >> 2);
    unsigned p1   = p0 + (1u << (s - 1));
    const float* A  = factors + ((size_t)s * 2048 + p0) * 4;
    const float* Bf = factors + ((size_t)s * 2048 + p1) * 4;
    float a00 = A[0],  a01 = A[1],  a10 = A[2],  a11 = A[3];
    float b00 = Bf[0], b01 = Bf[1], b10 = Bf[2], b11 = Bf[3];
    float nw[4][4];
    #pragma unroll
    for (int i = 0; i < 4; ++i) {
      nw[i][0] = w[i][0] * a00 + w[i][2] * a10;
      nw[i][2] = w[i][0] * a01 + w[i][2] * a11;
      nw[i][1] = w[i][1] * b00 + w[i][3] * b10;
      nw[i][3] = w[i][1] * b01 + w[i][3] * b11;
    }
    #pragma unroll
    for (int i = 0; i < 4; ++i)
      #pragma unroll
      for (int j = 0; j < 4; ++j) w[i][j] = nw[i][j];
  }

  float* o = Wfold + (size_t)g * 16;
  #pragma unroll
  for (int i = 0; i < 4; ++i)
    #pragma unroll
    for (int d = 0; d < 4; ++d) o[i * 4 + d] = w[i][d];
}

// ---------------------------------------------------------------------------
// Pass 2: out[t, 4g:4g+4] = x[t, 4g:4g+4] @ W_g + bias, via WMMA f32 16x16x4.
// Block = 256 threads (8 waves) = 128-column slab x 128 tokens.
// Each wave owns 16 output columns (4 groups -> 4 chained K=4 WMMAs/tile).
// X tiles are staged memory->LDS with CDNA5 async loads (ASYNCcnt).
// ---------------------------------------------------------------------------
__global__ __launch_bounds__(256) void bfly_wmma_kernel(
    const float* __restrict__ x, const float* __restrict__ Wfold,
    const float* __restrict__ bias, float* __restrict__ out) {
  __shared__ float tile[16 * LDS_PITCH];

  const int slab   = blockIdx.x;       // 0..31 : 128-column slab
  const int tchunk = blockIdx.y;       // 0..63 : 128-token chunk
  const int tid    = threadIdx.x;
  const int lane   = tid & 31;
  const int wave   = tid >> 5;
  const int l15    = lane & 15;
  const int lh     = lane >> 4;        // wave half: 0/1
  const int colbase = slab * 128 + wave * 16;   // wave's 16 output columns

  // Build the four B fragments (block-diagonal W_g) once per block.
  // B 4x16 layout: VGPR v, lanes 0-15 -> K = v; lanes 16-31 -> K = 2+v.
  v2f Bm[4];
  #pragma unroll
  for (int kk = 0; kk < 4; ++kk) {
    int gg = (colbase >> 2) + kk;
    float bx = 0.0f, by = 0.0f;
    if ((l15 >> 2) == kk) {            // only columns 4kk..4kk+3 are non-zero
      int d  = l15 & 3;
      int K0 = 2 * lh;
      bx = Wfold[(size_t)gg * 16 + (K0 + 0) * 4 + d];
      by = Wfold[(size_t)gg * 16 + (K0 + 1) * 4 + d];
    }
    Bm[kk].x = bx; Bm[kk].y = by;
  }
  const float bv = bias[colbase + l15];   // per-output-column bias (per lane)

  // Per-thread staging coordinates (2 float4 chunks of the 16x128 tile).
  const int row0 = tid >> 5,            cv0 = tid & 31;
  const int row1 = (tid + 256) >> 5,    cv1 = tid & 31;
  const unsigned lds0 =
      (unsigned)(uintptr_t)(tile + row0 * LDS_PITCH + cv0 * 4);
  const unsigned lds1 =
      (unsigned)(uintptr_t)(tile + row1 * LDS_PITCH + cv1 * 4);

  for (int t = 0; t < 8; ++t) {
    const int t0 = tchunk * 128 + t * 16;
    __syncthreads();   // previous tile fully consumed (ds reads retired)

    // CDNA5 async copy: memory -> LDS directly, tracked by ASYNCcnt.
    {
      unsigned long long ga0 =
          (unsigned long long)(uintptr_t)(x + (size_t)(t0 + row0) * BN +
                                          slab * 128 + cv0 * 4);
      unsigned long long ga1 =
          (unsigned long long)(uintptr_t)(x + (size_t)(t0 + row1) * BN +
                                          slab * 128 + cv1 * 4);
      asm volatile("global_load_async_to_lds_b128 %0, %1, off"
                   :: "v"(lds0), "v"(ga0) : "memory");
      asm volatile("global_load_async_to_lds_b128 %0, %1, off"
                   :: "v"(lds1), "v"(ga1) : "memory");
      if (t < 7) {  // warm next token tile (global_prefetch_b8)
        __builtin_prefetch((const char*)(uintptr_t)ga0 + (size_t)16 * BN * 4, 0, 1);
        __builtin_prefetch((const char*)(uintptr_t)ga1 + (size_t)16 * BN * 4, 0, 1);
      }
      // Compiler does not track inline-asm ASYNCcnt: wait explicitly before
      // the barrier so every wave's tile data is resident in LDS.
      asm volatile("s_wait_asynccnt 0x0" ::: "memory");
    }
    __syncthreads();

    v8f c = {0.f, 0.f, 0.f, 0.f, 0.f, 0.f, 0.f, 0.f};
    #pragma unroll
    for (int kk = 0; kk < 4; ++kk) {
      // A 16x4 layout: lane l holds M=l&15; VGPR v -> K = 2*(l>>4)+v.
      v2f a = *(const v2f*)(tile + l15 * LDS_PITCH + wave * 16 + kk * 4 + 2 * lh);
      c = __builtin_amdgcn_wmma_f32_16x16x4_f32(
          /*neg_a=*/false, a, /*neg_b=*/false, Bm[kk],
          /*c_mod=*/(short)0, c, /*reuse_a=*/false, /*reuse_b=*/false);
    }

    // D layout: VGPR i, lanes 0-15 -> (M=i, N=l15); lanes 16-31 -> (M=i+8).
    #pragma unroll
    for (int i = 0; i < 8; ++i) {
      out[(size_t)(t0 + i + 8 * lh) * BN + colbase + l15] = c[i] + bv;
    }
  }
}

extern "C" void kernel_launch(void* const* d_in, const int* in_sizes, int n_in,
                              void* d_out, int out_size, void* d_ws, size_t ws_size,
                              hipStream_t stream) {
  (void)in_sizes; (void)n_in; (void)out_size; (void)ws_size;
  const float* x       = (const float*)d_in[0];
  const float* factors = (const float*)d_in[1];
  const float* bias    = (const float*)d_in[2];
  float*       out     = (float*)d_out;
  float*       Wfold   = (float*)d_ws;   // 1024 * 16 floats = 64 KB

  bfly_fold_kernel<<<4, 256, 0, stream>>>(factors, Wfold);

  dim3 grid(BN / 128, BTOK / 128);       // 32 x 64 blocks
  bfly_wmma_kernel<<<grid, 256, 0, stream>>>(x, Wfold, bias, out);
}